// MacroDGRCL_19069654794982
// MI455X (gfx1250) — compile-verified
//
#include <hip/hip_runtime.h>
#include <math.h>

#define NN 4096
#define HIDDEN 128
#define HEAD_DIM 32
#define TOPK 15
#define EC (NN * 16)
#define NEG_SLOPE 0.2f

typedef float v2f __attribute__((ext_vector_type(2)));
typedef float v8f __attribute__((ext_vector_type(8)));

// ---------------------------------------------------------------------------
// Kernel 1: fold att into W:  Wc[c][col] ; col 0..3 = left heads, 4..7 = right
// heads, 8..15 = zero padding so B tiles are 16 wide for WMMA.
// ---------------------------------------------------------------------------
__global__ void build_wc(const float* __restrict__ W, const float* __restrict__ att,
                         float* __restrict__ Wc) {
  int idx = blockIdx.x * blockDim.x + threadIdx.x;   // 0 .. 2047
  if (idx >= HIDDEN * 16) return;
  int c = idx >> 4;
  int col = idx & 15;
  float s = 0.0f;
  if (col < 8) {
    int h = col & 3;
    int roff = (col >= 4) ? HEAD_DIM : 0;
    const float* w = W + c * HIDDEN + h * HEAD_DIM;
    const float* a = att + h * (2 * HEAD_DIM) + roff;
#pragma unroll 8
    for (int d = 0; d < HEAD_DIM; ++d) s += w[d] * a[d];
  }
  Wc[c * 16 + col] = s;
}

// ---------------------------------------------------------------------------
// Kernel 2: LR[4096][8] = E[4096][128] @ Wc[128][16] (cols 8..15 discarded)
// via v_wmma_f32_16x16x4_f32. One wave per 16-row tile, K unrolled -> 32 WMMA.
// A layout (16x4 f32): lanes 0-15 M=0..15 {K0,K1}, lanes 16-31 M=0..15 {K2,K3}.
// C/D layout: vgpr r -> (M = r + 8*(lane>=16), N = lane%16).
// ---------------------------------------------------------------------------
__global__ __launch_bounds__(32) void lr_wmma(const float* __restrict__ E,
                                              const float* __restrict__ Wc,
                                              float* __restrict__ LR) {
  int tile = blockIdx.x;            // 256 tiles of 16 rows
  int lane = threadIdx.x;           // 0..31 (wave32)
  int half = lane >> 4;             // 0: K0/K1, 1: K2/K3
  int m = lane & 15;

  const float* Arow = E + (size_t)(tile * 16 + m) * HIDDEN;
  v8f c = {0.f, 0.f, 0.f, 0.f, 0.f, 0.f, 0.f, 0.f};

#pragma unroll
  for (int k0 = 0; k0 < HIDDEN; k0 += 4) {
    int k = k0 + half * 2;
    float2 av = *(const float2*)(Arow + k);   // 8B aligned
    v2f a, b;
    a.x = av.x;
    a.y = av.y;
    b.x = Wc[(k + 0) * 16 + m];
    b.y = Wc[(k + 1) * 16 + m];
    c = __builtin_amdgcn_wmma_f32_16x16x4_f32(false, a, false, b, (short)0, c,
                                              false, false);
  }

  if (m < 8) {
#pragma unroll
    for (int r = 0; r < 8; ++r) {
      LR[(size_t)(tile * 16 + r + half * 8) * 8 + m] = c[r];
    }
  }
}

// ---------------------------------------------------------------------------
// Kernels 3-6: build per-row CSR of correlation edges.
// ---------------------------------------------------------------------------
__global__ void zero_i32(int* __restrict__ p, int n) {
  int i = blockIdx.x * blockDim.x + threadIdx.x;
  if (i < n) p[i] = 0;
}

__global__ void count_rows(const int* __restrict__ eidx, int* __restrict__ counts) {
  int e = blockIdx.x * blockDim.x + threadIdx.x;
  if (e < EC) atomicAdd(&counts[eidx[e]], 1);
}

__global__ __launch_bounds__(256) void scan4096(const int* __restrict__ counts,
                                                int* __restrict__ offsets) {
  __shared__ int part[256];
  int t = threadIdx.x;
  int base = t * 16;
  int local[16];
  int sum = 0;
#pragma unroll
  for (int k = 0; k < 16; ++k) { local[k] = sum; sum += counts[base + k]; }
  part[t] = sum;
  __syncthreads();
  for (int off = 1; off < 256; off <<= 1) {
    int v = (t >= off) ? part[t - off] : 0;
    __syncthreads();
    part[t] += v;
    __syncthreads();
  }
  int prev = (t > 0) ? part[t - 1] : 0;
#pragma unroll
  for (int k = 0; k < 16; ++k) offsets[base + k] = prev + local[k];
}

__global__ void scatter_edges(const int* __restrict__ eidx, const float* __restrict__ ew,
                              const int* __restrict__ offsets, int* __restrict__ fill,
                              int* __restrict__ ecols, float* __restrict__ ewsort) {
  int e = blockIdx.x * blockDim.x + threadIdx.x;
  if (e < EC) {
    int r = eidx[e];
    int pos = offsets[r] + atomicAdd(&fill[r], 1);
    ecols[pos] = eidx[EC + e];
    ewsort[pos] = ew[e];
  }
}

// ---------------------------------------------------------------------------
// Kernel 7: per-row scores + top-15 + softmax. One wave per row, 2 waves/block,
// each wave owns a private 16KB LDS slice (dense corr row, then merge buffer).
// ---------------------------------------------------------------------------
__global__ __launch_bounds__(64) void topk_rows(
    const float* __restrict__ LR, const float* __restrict__ lamp,
    const unsigned char* __restrict__ act, const int* __restrict__ counts,
    const int* __restrict__ offsets, const int* __restrict__ ecols,
    const float* __restrict__ ewsort, float* __restrict__ out) {
  __shared__ float sh[2 * NN];
  const int wid = threadIdx.x >> 5;
  const int lane = threadIdx.x & 31;
  const int row = blockIdx.x * 2 + wid;
  float* cbuf = sh + wid * NN;

  // dense corr row in LDS (overwrite semantics matches jnp .set with dups)
  for (int j = lane; j < NN; j += 32) cbuf[j] = 0.0f;
  int beg = offsets[row];
  int cnt = counts[row];
  for (int t = lane; t < cnt; t += 32) cbuf[ecols[beg + t]] = ewsort[beg + t];
  __syncthreads();

  const float lam = lamp[0];
  const bool acti = act[row] != 0;
  const float l0 = LR[row * 8 + 0];
  const float l1 = LR[row * 8 + 1];
  const float l2 = LR[row * 8 + 2];
  const float l3 = LR[row * 8 + 3];
  const float NINF = -__builtin_inff();

  float s[TOPK];
  int id[TOPK];
#pragma unroll
  for (int t = 0; t < TOPK; ++t) { s[t] = NINF; id[t] = 0x7fffffff; }

  for (int j = lane; j < NN; j += 32) {
    const float4 r = *(const float4*)(LR + (size_t)j * 8 + 4);
    float a0 = l0 + r.x, a1 = l1 + r.y, a2 = l2 + r.z, a3 = l3 + r.w;
    float v = fmaxf(a0, NEG_SLOPE * a0) + fmaxf(a1, NEG_SLOPE * a1) +
              fmaxf(a2, NEG_SLOPE * a2) + fmaxf(a3, NEG_SLOPE * a3);
    v *= 0.25f;
    bool ok = acti && (act[j] != 0);
    v = ok ? (v + lam * cbuf[j]) : NINF;
    if (v > s[TOPK - 1]) {         // guarded insertion, arrays stay in VGPRs
      float cv = v;
      int ci = j;
#pragma unroll
      for (int t = 0; t < TOPK; ++t) {
        float os = s[t];
        int oi = id[t];
        bool g = cv > os;
        s[t] = g ? cv : os;
        id[t] = g ? ci : oi;
        cv = g ? os : cv;
        ci = g ? oi : ci;
      }
    }
  }

  // dump lane-local sorted lists to LDS, then 15-round cross-lane selection
  int* ibuf = (int*)(cbuf + 512);
#pragma unroll
  for (int t = 0; t < TOPK; ++t) {
    cbuf[lane * TOPK + t] = s[t];
    ibuf[lane * TOPK + t] = id[t];
  }
  __syncthreads();

  int p = 0;
  float out_s = NINF;
  int out_j = 0x7fffffff;
  for (int r = 0; r < TOPK; ++r) {
    float hs = (p < TOPK) ? cbuf[lane * TOPK + p] : NINF;
    int hj = (p < TOPK) ? ibuf[lane * TOPK + p] : 0x7fffffff;
    float bs = hs;
    int bj = hj;
    int bl = lane;
#pragma unroll
    for (int off = 16; off >= 1; off >>= 1) {
      float os = __shfl_xor(bs, off, 32);
      int oj = __shfl_xor(bj, off, 32);
      int ol = __shfl_xor(bl, off, 32);
      bool take = (os > bs) || (os == bs && oj < bj);   // low index wins ties
      bs = take ? os : bs;
      bj = take ? oj : bj;
      bl = take ? ol : bl;
    }
    if (lane == bl) p++;
    if (lane == r) { out_s = bs; out_j = bj; }
  }

  // softmax over the 15 kept scores (lane 0 holds the max)
  float mx = __shfl(out_s, 0, 32);
  float e = (lane < TOPK) ? __expf(out_s - mx) : 0.0f;
  float sum = e;
#pragma unroll
  for (int off = 16; off >= 1; off >>= 1) sum += __shfl_xor(sum, off, 32);

  if (lane < TOPK) {
    int o = row * TOPK + lane;
    int* oi = (int*)out;
    oi[o] = out_j;                 // edge_index[0]: source (col) ids
    oi[NN * TOPK + o] = row;       // edge_index[1]: row ids
    out[2 * NN * TOPK + o] = e / sum;  // edge_weight
  }
}

// ---------------------------------------------------------------------------
extern "C" void kernel_launch(void* const* d_in, const int* in_sizes, int n_in,
                              void* d_out, int out_size, void* d_ws, size_t ws_size,
                              hipStream_t stream) {
  const float* emb = (const float*)d_in[0];          // [4096,128]
  const float* W = (const float*)d_in[1];            // [128,128]
  const float* att = (const float*)d_in[2];          // [4,64]
  const float* lam = (const float*)d_in[3];          // [1]
  const unsigned char* act = (const unsigned char*)d_in[4];  // [4096] bool
  const int* eidx = (const int*)d_in[5];             // [2,65536]
  const float* ew = (const float*)d_in[6];           // [65536]
  float* out = (float*)d_out;

  char* ws = (char*)d_ws;
  float* Wc = (float*)(ws);                          //   8 KB
  float* LR = (float*)(ws + 8192);                   // 128 KB
  int* counts = (int*)(ws + 8192 + 131072);          //  16 KB
  int* fill = counts + NN;                           //  16 KB
  int* offsets = fill + NN;                          //  16 KB
  int* ecols = offsets + NN;                         // 256 KB
  float* ewsort = (float*)(ecols + EC);              // 256 KB

  build_wc<<<8, 256, 0, stream>>>(W, att, Wc);
  lr_wmma<<<256, 32, 0, stream>>>(emb, Wc, LR);
  zero_i32<<<32, 256, 0, stream>>>(counts, 2 * NN);  // counts + fill
  count_rows<<<256, 256, 0, stream>>>(eidx, counts);
  scan4096<<<1, 256, 0, stream>>>(counts, offsets);
  scatter_edges<<<256, 256, 0, stream>>>(eidx, ew, offsets, fill, ecols, ewsort);
  topk_rows<<<2048, 64, 0, stream>>>(LR, lam, act, counts, offsets, ecols, ewsort, out);
}